// XhatModule_7198365188377
// MI455X (gfx1250) — compile-verified
//
#include <hip/hip_runtime.h>
#include <math.h>

// ---------------------------------------------------------------------------
// Batched biexponential Gauss-Newton (softplus-reparameterized, Armijo LS).
// Mapping: one wave32 per curve -> wave-uniform control flow (no divergence
// from per-curve iteration counts). Lanes split T=256 into 8 points each.
// y staged global->LDS via CDNA5 async-LDS ops, then held in VGPRs.
// ---------------------------------------------------------------------------

#define WAVES_PER_BLOCK 8
#define TPTS 256
#define PPL 8            // points per lane = 256/32

#define GN_MAX_ITER 300
#define GN_LS_MAX   25
#define GN_TOL      1e-9f
#define GN_ARMIJO   1e-4f
#define GN_JITTER   1e-7f

#define GLOBAL_AS __attribute__((address_space(1)))
#define LDS_AS    __attribute__((address_space(3)))

typedef int v4i __attribute__((ext_vector_type(4)));

#if defined(__HIP_DEVICE_COMPILE__) && defined(__gfx1250__) && \
    __has_builtin(__builtin_amdgcn_global_load_async_to_lds_b128)
#define USE_ASYNC_LDS 1
#else
#define USE_ASYNC_LDS 0
#endif

__device__ __forceinline__ float wave_sum32(float v) {
  // Butterfly all-reduce across the 32 lanes of a CDNA5 wave.
  v += __shfl_xor(v, 1, 32);
  v += __shfl_xor(v, 2, 32);
  v += __shfl_xor(v, 4, 32);
  v += __shfl_xor(v, 8, 32);
  v += __shfl_xor(v, 16, 32);
  return v;
}

__device__ __forceinline__ float softplus_f(float u) {
  // jax.nn.softplus: max(u,0) + log1p(exp(-|u|)) (stable form)
  return fmaxf(u, 0.0f) + log1pf(__expf(-fabsf(u)));
}
__device__ __forceinline__ float sigmoid_f(float u) {
  return 1.0f / (1.0f + __expf(-u));
}
__device__ __forceinline__ float inv_sp_f(float x) {
  // stable inverse softplus for positive x
  x = fmaxf(x, 1e-6f);
  return x + logf(-expm1f(-x));
}

__global__ __launch_bounds__(256)
void biexp_gn_kernel(const float* __restrict__ log_mu,
                     const float* __restrict__ y,
                     const float* __restrict__ t,
                     const float* __restrict__ x_init,
                     float* __restrict__ out, int B) {
  __shared__ float sy[WAVES_PER_BLOCK * TPTS];   // 8 KB

  const int wave = (int)(threadIdx.x >> 5);
  const int lane = (int)(threadIdx.x & 31);
  int curve = (int)blockIdx.x * WAVES_PER_BLOCK + wave;
  const bool valid = (curve < B);
  if (curve >= B) curve = B - 1;   // clamp; keep wave alive (no barriers below)

  // ---- Stage this wave's curve (1 KB of y) into LDS ----------------------
#if USE_ASYNC_LDS
  {
    const GLOBAL_AS float* gp =
        (const GLOBAL_AS float*)(y + (size_t)curve * TPTS) + lane * 4;
    LDS_AS float* lp = (LDS_AS float*)(&sy[wave * TPTS]) + lane * 4;
    // 32 lanes x 16B = 512B per op; two ops cover 1KB. INST_OFFSET is added
    // to both the global and LDS addresses (ISA 15.18.3 async pseudocode).
    __builtin_amdgcn_global_load_async_to_lds_b128((GLOBAL_AS v4i*)gp,
                                                   (LDS_AS v4i*)lp, 0, 0);
    __builtin_amdgcn_global_load_async_to_lds_b128((GLOBAL_AS v4i*)gp,
                                                   (LDS_AS v4i*)lp, 512, 0);
#if __has_builtin(__builtin_amdgcn_s_wait_asynccnt)
    __builtin_amdgcn_s_wait_asynccnt(0);
#else
    asm volatile("s_wait_asynccnt 0x0" ::: "memory");
#endif
    asm volatile("" ::: "memory");  // LDS now holds async-written data
  }
#else
  {
    const float* gp = y + (size_t)curve * TPTS + lane * 4;
    float4 a = *(const float4*)gp;
    float4 b = *(const float4*)(gp + 128);
    *(float4*)(&sy[wave * TPTS + lane * 4]) = a;
    *(float4*)(&sy[wave * TPTS + 128 + lane * 4]) = b;
    __syncthreads();
  }
#endif

  // ---- Pull per-lane samples into registers; everything below is in-reg --
  float ty[PPL], tt[PPL];
#pragma unroll
  for (int j = 0; j < PPL; ++j) {
    const int idx = lane + 32 * j;          // stride-32: conflict-free banks
    tt[j] = t[idx];
    ty[j] = sy[wave * TPTS + idx];
  }

  const float mu = __expf(log_mu[0]);
  float u[4];
#pragma unroll
  for (int a = 0; a < 4; ++a)
    u[a] = inv_sp_f(fmaxf(x_init[a], 1e-3f));

  bool done = false;
  for (int it = 0; it < GN_MAX_ITER && !done; ++it) {
    float x[4], sd[4];
#pragma unroll
    for (int a = 0; a < 4; ++a) { x[a] = softplus_f(u[a]); sd[a] = sigmoid_f(u[a]); }

    // Gram of [J_u | r]: 10 upper-tri H entries + 4 J^T r + r^T r
    float h00 = 0.f, h01 = 0.f, h02 = 0.f, h03 = 0.f;
    float h11 = 0.f, h12 = 0.f, h13 = 0.f;
    float h22 = 0.f, h23 = 0.f, h33 = 0.f;
    float g0 = 0.f, g1 = 0.f, g2 = 0.f, g3 = 0.f, rr = 0.f;
#pragma unroll
    for (int j = 0; j < PPL; ++j) {
      const float e0 = __expf(-x[2] * tt[j]);
      const float e1 = __expf(-x[3] * tt[j]);
      const float r  = ty[j] - fmaf(x[0], e0, x[1] * e1);
      const float j0 = e0 * sd[0];
      const float j1 = e1 * sd[1];
      const float j2 = -x[0] * tt[j] * e0 * sd[2];
      const float j3 = -x[1] * tt[j] * e1 * sd[3];
      h00 = fmaf(j0, j0, h00); h01 = fmaf(j0, j1, h01);
      h02 = fmaf(j0, j2, h02); h03 = fmaf(j0, j3, h03);
      h11 = fmaf(j1, j1, h11); h12 = fmaf(j1, j2, h12);
      h13 = fmaf(j1, j3, h13); h22 = fmaf(j2, j2, h22);
      h23 = fmaf(j2, j3, h23); h33 = fmaf(j3, j3, h33);
      g0 = fmaf(j0, r, g0); g1 = fmaf(j1, r, g1);
      g2 = fmaf(j2, r, g2); g3 = fmaf(j3, r, g3);
      rr = fmaf(r, r, rr);
    }
    h00 = wave_sum32(h00); h01 = wave_sum32(h01); h02 = wave_sum32(h02);
    h03 = wave_sum32(h03); h11 = wave_sum32(h11); h12 = wave_sum32(h12);
    h13 = wave_sum32(h13); h22 = wave_sum32(h22); h23 = wave_sum32(h23);
    h33 = wave_sum32(h33);
    g0 = wave_sum32(g0); g1 = wave_sum32(g1);
    g2 = wave_sum32(g2); g3 = wave_sum32(g3);
    rr = wave_sum32(rr);

    const float xs = x[0]*x[0] + x[1]*x[1] + x[2]*x[2] + x[3]*x[3];
    const float loss = rr + mu * xs;
    float grad[4] = { -g0 + mu * x[0] * sd[0], -g1 + mu * x[1] * sd[1],
                      -g2 + mu * x[2] * sd[2], -g3 + mu * x[3] * sd[3] };
    const float dg = mu + GN_JITTER;

    // Augmented SPD 4x5 system [H | -grad]; pivot-free GE (H >= mu*I).
    float A[4][5] = {
      { h00 + dg, h01, h02, h03, -grad[0] },
      { h01, h11 + dg, h12, h13, -grad[1] },
      { h02, h12, h22 + dg, h23, -grad[2] },
      { h03, h13, h23, h33 + dg, -grad[3] } };
#pragma unroll
    for (int k = 0; k < 4; ++k) {
      const float inv = 1.0f / A[k][k];
#pragma unroll
      for (int c = k; c < 5; ++c) A[k][c] *= inv;
#pragma unroll
      for (int i2 = k + 1; i2 < 4; ++i2) {
        const float f = A[i2][k];
#pragma unroll
        for (int c = k; c < 5; ++c) A[i2][c] = fmaf(-f, A[k][c], A[i2][c]);
      }
    }
    float du[4];
    du[3] = A[3][4];
    du[2] = A[2][4] - A[2][3] * du[3];
    du[1] = A[1][4] - A[1][2] * du[2] - A[1][3] * du[3];
    du[0] = A[0][4] - A[0][1] * du[1] - A[0][2] * du[2] - A[0][3] * du[3];

    const float gd = grad[0]*du[0] + grad[1]*du[1] + grad[2]*du[2] + grad[3]*du[3];

    // Armijo backtracking (wave-uniform loop)
    float step = 1.0f;
    bool ok = false;
    for (int k = 0; k < GN_LS_MAX && !ok; ++k) {
      float xn[4];
#pragma unroll
      for (int a = 0; a < 4; ++a) xn[a] = softplus_f(fmaf(step, du[a], u[a]));
      float rs = 0.f;
#pragma unroll
      for (int j = 0; j < PPL; ++j) {
        const float e0 = __expf(-xn[2] * tt[j]);
        const float e1 = __expf(-xn[3] * tt[j]);
        const float r  = ty[j] - fmaf(xn[0], e0, xn[1] * e1);
        rs = fmaf(r, r, rs);
      }
      rs = wave_sum32(rs);
      const float ln = rs + mu * (xn[0]*xn[0] + xn[1]*xn[1] + xn[2]*xn[2] + xn[3]*xn[3]);
      ok = (ln <= loss - GN_ARMIJO * step * gd);
      if (!ok) step *= 0.5f;
    }

    const float dn2 = du[0]*du[0] + du[1]*du[1] + du[2]*du[2] + du[3]*du[3];
#pragma unroll
    for (int a = 0; a < 4; ++a) u[a] = fmaf(step, du[a], u[a]);
    done = (step * sqrtf(dn2) < GN_TOL);
  }

  if (valid && lane == 0) {
    float4 r;
    r.x = softplus_f(u[0]); r.y = softplus_f(u[1]);
    r.z = softplus_f(u[2]); r.w = softplus_f(u[3]);
    *(float4*)(out + (size_t)curve * 4) = r;
  }
}

extern "C" void kernel_launch(void* const* d_in, const int* in_sizes, int n_in,
                              void* d_out, int out_size, void* d_ws, size_t ws_size,
                              hipStream_t stream) {
  const float* log_mu = (const float*)d_in[0];   // scalar
  const float* y      = (const float*)d_in[1];   // [B, 256]
  const float* t      = (const float*)d_in[2];   // [256]
  const float* x_init = (const float*)d_in[3];   // [4]
  float* out = (float*)d_out;                    // [B, 4]
  (void)d_ws; (void)ws_size; (void)n_in; (void)out_size;

  const int B = in_sizes[1] / TPTS;
  const int blocks = (B + WAVES_PER_BLOCK - 1) / WAVES_PER_BLOCK;
  hipLaunchKernelGGL(biexp_gn_kernel, dim3(blocks), dim3(256), 0, stream,
                     log_mu, y, t, x_init, out, B);
}